// GCNAggregator_15745350107506
// MI455X (gfx1250) — compile-verified
//
#include <hip/hip_runtime.h>
#include <stdint.h>

#ifndef __has_builtin
#define __has_builtin(x) 0
#endif

// The gfx1250 async-copy builtin wants GCC-vector int4 pointers with explicit
// address spaces: param0 = int4 addrspace(1)* (global), param1 = int4
// addrspace(3)* (LDS), then imm offset + imm cpol.
typedef int v4i __attribute__((vector_size(16)));
typedef __attribute__((address_space(1))) v4i as1_v4i;
typedef __attribute__((address_space(3))) v4i as3_v4i;

// --- CDNA5 async global->LDS copy (16 bytes per lane), with fallback -------
__device__ __forceinline__ void async_gather_b128(const float* gsrc, float* ldst) {
#if __has_builtin(__builtin_amdgcn_global_load_async_to_lds_b128)
    __builtin_amdgcn_global_load_async_to_lds_b128(
        (as1_v4i*)gsrc, (as3_v4i*)ldst, 0, 0);
#else
    // Synchronous fallback: VGPR round-trip copy.
    float4 v = *(const float4*)gsrc;
    *(float4*)ldst = v;
#endif
}

__device__ __forceinline__ void wait_async_zero() {
#if __has_builtin(__builtin_amdgcn_s_wait_asynccnt)
    __builtin_amdgcn_s_wait_asynccnt(0);
#else
#if defined(__AMDGCN__)
    asm volatile("s_wait_asynccnt 0" ::: "memory");
#endif
#endif
}

// ---------------------------------------------------------------------------
// Kernel 1: zero the column-degree counters (d_ws is poisoned, must clear).
__global__ __launch_bounds__(256) void k_zero_u32(unsigned* __restrict__ p, int n) {
    int i = blockIdx.x * blockDim.x + threadIdx.x;
    if (i < n) p[i] = 0u;
}

// ---------------------------------------------------------------------------
// Kernel 2: one wave32 per row. Dedup the 32 neighbor ids within the row,
// emit keep-bitmask + 1/sqrt(row_degree), and bump distinct column counters.
__global__ __launch_bounds__(128) void k_degrees(
    const int* __restrict__ neigh, unsigned* __restrict__ colcnt,
    unsigned* __restrict__ maskbits, float* __restrict__ inv_row, int nrows) {
    const int lane = threadIdx.x & 31;
    const int row  = blockIdx.x * (blockDim.x >> 5) + (threadIdx.x >> 5);
    if (row >= nrows) return;

    const int c = neigh[row * 32 + lane];
    bool dup = false;
#pragma unroll
    for (int j = 0; j < 32; ++j) {
        int cj = __shfl(c, j, 32);
        if (j < lane && cj == c) dup = true;
    }
    const unsigned keep = (unsigned)(__ballot(!dup) & 0xffffffffull);
    if (!dup) atomicAdd(&colcnt[c], 1u);
    if (lane == 0) {
        maskbits[row] = keep;
        inv_row[row]  = rsqrtf((float)__popc(keep));   // row degree >= 1
    }
}

// ---------------------------------------------------------------------------
// Kernel 3: inv_col[u] = 1 / max(sqrt(cnt), 1).
__global__ __launch_bounds__(256) void k_invcol(
    const unsigned* __restrict__ colcnt, float* __restrict__ inv_col, int n) {
    int i = blockIdx.x * blockDim.x + threadIdx.x;
    if (i < n) {
        unsigned c = colcnt[i];
        inv_col[i] = c ? rsqrtf((float)c) : 1.0f;
    }
}

// ---------------------------------------------------------------------------
// Kernel 4: one 128-thread block (4 waves) per output row.
//  - threads 0..31 resolve per-slot weight and embedding-row base address
//  - each wave issues 8 async B128 gathers: 32 lanes x 16 B = one full
//    512-byte embedding row per instruction, all 32 rows staged in LDS
//  - s_wait_asynccnt 0 + barrier, then a 32-term FMA reduction per dim.
__global__ __launch_bounds__(128) void k_aggregate(
    const int* __restrict__ neigh, const int* __restrict__ uids,
    const float* __restrict__ emb, const float* __restrict__ inv_col,
    const float* __restrict__ inv_row, const unsigned* __restrict__ maskbits,
    float* __restrict__ out) {
    __shared__ float    s_rows[32 * 128];       // 16 KB staged embedding rows
    __shared__ float    s_w[32];                // per-slot weight (0 for dups)
    __shared__ uint64_t s_ga[32];               // per-slot global row address

    const int b   = blockIdx.x;
    const int tid = threadIdx.x;

    if (tid < 32) {
        const unsigned mb = maskbits[b];
        const int c  = neigh[b * 32 + tid];
        s_w[tid]  = ((mb >> tid) & 1u) ? inv_col[c] : 0.0f;
        const long long id = uids[c];
        s_ga[tid] = (uint64_t)(const void*)(emb + id * 128);
    }
    __syncthreads();

    const int lane = tid & 31;
    const int wave = tid >> 5;
#pragma unroll
    for (int i = 0; i < 8; ++i) {
        const int r = wave * 8 + i;
        const float* gsrc = (const float*)(s_ga[r] + (uint64_t)(lane * 16));
        async_gather_b128(gsrc, &s_rows[r * 128 + lane * 4]);
    }
    wait_async_zero();
    __syncthreads();

    float acc = 0.0f;
#pragma unroll
    for (int r = 0; r < 32; ++r)
        acc = fmaf(s_w[r], s_rows[r * 128 + tid], acc);

    out[b * 128 + tid] = inv_row[b] * acc;
}

// ---------------------------------------------------------------------------
extern "C" void kernel_launch(void* const* d_in, const int* in_sizes, int n_in,
                              void* d_out, int out_size, void* d_ws, size_t ws_size,
                              hipStream_t stream) {
    const int*   neigh = (const int*)d_in[0];   // [B, 32] int32
    const int*   uids  = (const int*)d_in[1];   // [U] int32
    const float* emb   = (const float*)d_in[2]; // [V, 128] float32
    float*       out   = (float*)d_out;         // [B, 128] float32

    const int B = in_sizes[0] / 32;
    const int U = in_sizes[1];

    // Workspace layout (all 4-byte aligned):
    //   [0, 4U)            colcnt   (u32)
    //   [4U, 8U)           inv_col  (f32)
    //   [8U, 8U+4B)        inv_row  (f32)
    //   [8U+4B, 8U+8B)     maskbits (u32)
    uint8_t*  ws       = (uint8_t*)d_ws;
    unsigned* colcnt   = (unsigned*)(ws);
    float*    inv_col  = (float*)(ws + (size_t)U * 4);
    float*    inv_row  = (float*)(ws + (size_t)U * 8);
    unsigned* maskbits = (unsigned*)(ws + (size_t)U * 8 + (size_t)B * 4);

    k_zero_u32 <<<(U + 255) / 256, 256, 0, stream>>>(colcnt, U);
    k_degrees  <<<(B + 3) / 4,     128, 0, stream>>>(neigh, colcnt, maskbits, inv_row, B);
    k_invcol   <<<(U + 255) / 256, 256, 0, stream>>>(colcnt, inv_col, U);
    k_aggregate<<<B,               128, 0, stream>>>(neigh, uids, emb, inv_col,
                                                     inv_row, maskbits, out);
}